// GatedGCNLayer_23759759082192
// MI455X (gfx1250) — compile-verified
//
#include <hip/hip_runtime.h>
#include <hip/hip_bf16.h>

#define DD 64
#define NN 80000
#define EE 1280000
#define ROWP 68            // padded LDS row (floats) -> conflict-free frag reads
#define TILEF (16 * ROWP)  // floats per staged tile buffer

typedef __attribute__((ext_vector_type(2))) float v2f;
typedef __attribute__((ext_vector_type(4))) float v4f;
typedef __attribute__((ext_vector_type(8))) float v8f;

__device__ __forceinline__ v8f wmma_f32(v2f a, v2f b, v8f c) {
  // D = A(16x4 f32) * B(4x16 f32) + C(16x16 f32)
  return __builtin_amdgcn_wmma_f32_16x16x4_f32(
      false, a, false, b, (short)0, c, false, false);
}

__device__ __forceinline__ float sigmoidf_fast(float z) {
  return 1.0f / (1.0f + __expf(-z));
}

__device__ __forceinline__ uint32_t lds_off(const void* p) {
  // generic pointer to LDS: low 32 bits are the wave-relative LDS byte offset
  return (uint32_t)(uintptr_t)p;
}

// Issue async DMA of one 16x64 f32 tile (4 KB) into a padded LDS buffer.
// 8 x b128, each fully coalesced (32 lanes x 16B contiguous).
__device__ __forceinline__ void stage_tile_async(const float* __restrict__ gbase,
                                                 uint32_t lbase, int lane) {
#pragma unroll
  for (int i = 0; i < 8; ++i) {
    const int g = i * 32 + lane;       // float4 group 0..255
    const int row = g >> 4, c4 = g & 15;
    uint32_t la = lbase + (uint32_t)(row * (ROWP * 4) + c4 * 16);
    const float* ga = gbase + (size_t)g * 4;
    asm volatile("global_load_async_to_lds_b128 %0, %1, off"
                 :: "v"(la), "v"(ga) : "memory");
  }
}

// ---------------------------------------------------------------- K0: zero
__global__ void __launch_bounds__(256) zero_ws_k(float* __restrict__ p, size_t n) {
  size_t i = (size_t)blockIdx.x * blockDim.x + threadIdx.x;
  size_t stride = (size_t)gridDim.x * blockDim.x;
  for (; i < n; i += stride) p[i] = 0.0f;
}

// ---------------------------------------------------- K1: node GEMMs (WMMA)
__global__ void __launch_bounds__(256)
node_gemm4_k(const float* __restrict__ x,
             const float* __restrict__ WA, const float* __restrict__ bA,
             const float* __restrict__ WB, const float* __restrict__ bB,
             const float* __restrict__ WDm, const float* __restrict__ bD,
             const float* __restrict__ WEm, const float* __restrict__ bE,
             float* __restrict__ Ax, float* __restrict__ Bx,
             float* __restrict__ Dx, float* __restrict__ Ex) {
  __shared__ float wt[4][DD * DD];  // transposed: wt[w][n*64 + k]
  const float* Ws[4] = {WA, WB, WDm, WEm};
  const float* bs[4] = {bA, bB, bD, bE};
  float* outs[4] = {Ax, Bx, Dx, Ex};
  const int t = threadIdx.x;
  {
    int n = t >> 2;
    int kb = (t & 3) * 16;
#pragma unroll
    for (int w = 0; w < 4; ++w)
#pragma unroll
      for (int j = 0; j < 16; ++j)
        wt[w][n * DD + kb + j] = Ws[w][(kb + j) * DD + n];
  }
  __syncthreads();

  const int wid = t >> 5, lane = t & 31;
  const int mlo = lane & 15, hv = lane >> 4, kh = hv * 2;
  const int tile = blockIdx.x * 8 + wid;  // grid = 625 -> tiles 0..4999
  const int m0 = tile * 16;
  const v8f vzero = {0.f, 0.f, 0.f, 0.f, 0.f, 0.f, 0.f, 0.f};

  v2f af[16];
  const float* xr = x + (size_t)(m0 + mlo) * DD + kh;
#pragma unroll
  for (int kk = 0; kk < 16; ++kk) af[kk] = *(const v2f*)(xr + kk * 4);

#pragma unroll
  for (int w = 0; w < 4; ++w) {
    v2f bf[16][4];
#pragma unroll
    for (int kk = 0; kk < 16; ++kk)
#pragma unroll
      for (int nt = 0; nt < 4; ++nt)
        bf[kk][nt] = *(const v2f*)&wt[w][(nt * 16 + mlo) * DD + kk * 4 + kh];

    v8f acc[4];
#pragma unroll
    for (int nt = 0; nt < 4; ++nt) acc[nt] = vzero;
#pragma unroll
    for (int kk = 0; kk < 16; ++kk)
#pragma unroll
      for (int nt = 0; nt < 4; ++nt)
        acc[nt] = wmma_f32(af[kk], bf[kk][nt], acc[nt]);

    float* op = outs[w] + (size_t)(m0 + hv * 8) * DD + mlo;
#pragma unroll
    for (int nt = 0; nt < 4; ++nt) {
      const float bias = bs[w][nt * 16 + mlo];
#pragma unroll
      for (int r = 0; r < 8; ++r) op[(size_t)r * DD + nt * 16] = acc[nt][r] + bias;
    }
  }
}

// ------------------------------------------- K2: edge pass 1 (Ce GEMM + agg)
// 128 threads = 4 waves; each wave owns a double-buffered LDS staging area.
__global__ void __launch_bounds__(128)
edge_pass1_k(const float* __restrict__ e, const int* __restrict__ src,
             const int* __restrict__ dst, const float* __restrict__ WC,
             const float* __restrict__ bC, const float* __restrict__ Bx,
             const float* __restrict__ Dx, const float* __restrict__ Ex,
             float* __restrict__ num, float* __restrict__ den,
             float* __restrict__ stats) {
  __shared__ __align__(16) float etile[4][2][TILEF];
  __shared__ float wt[DD * DD];
  __shared__ float sstat[2 * DD];
  const int t = threadIdx.x;
  {
    int n = t >> 1, kb = (t & 1) * 32;
#pragma unroll
    for (int j = 0; j < 32; ++j) wt[n * DD + kb + j] = WC[(kb + j) * DD + n];
  }
  sstat[t] = 0.0f;
  __syncthreads();

  const int wid = t >> 5, lane = t & 31;
  const int mlo = lane & 15, hv = lane >> 4, kh = hv * 2;
  const v8f vzero = {0.f, 0.f, 0.f, 0.f, 0.f, 0.f, 0.f, 0.f};

  // hoist B fragments + bias out of the tile loop
  v2f bf[16][4];
#pragma unroll
  for (int kk = 0; kk < 16; ++kk)
#pragma unroll
    for (int nt = 0; nt < 4; ++nt)
      bf[kk][nt] = *(const v2f*)&wt[(nt * 16 + mlo) * DD + kk * 4 + kh];
  float cb[4];
#pragma unroll
  for (int nt = 0; nt < 4; ++nt) cb[nt] = bC[nt * 16 + mlo];

  const float* p0 = &etile[wid][0][0];
  const float* p1 = &etile[wid][1][0];
  const uint32_t l0 = lds_off(p0), l1 = lds_off(p1);

  const int nTiles = EE / 16;
  const int step = gridDim.x * 4;
  int tile = blockIdx.x * 4 + wid;
  int b = 0;
  if (tile < nTiles) stage_tile_async(e + (size_t)tile * 16 * DD, l0, lane);
  for (; tile < nTiles; tile += step, b ^= 1) {
    const int nxt = tile + step;
    if (nxt < nTiles) {
      stage_tile_async(e + (size_t)nxt * 16 * DD, b ? l0 : l1, lane);
      asm volatile("s_wait_asynccnt 8" ::: "memory");
    } else {
      asm volatile("s_wait_asynccnt 0" ::: "memory");
    }
    const float* my = b ? p1 : p0;

    v2f af[16];
#pragma unroll
    for (int kk = 0; kk < 16; ++kk)
      af[kk] = *(const v2f*)&my[mlo * ROWP + kk * 4 + kh];

    v8f acc[4];
#pragma unroll
    for (int nt = 0; nt < 4; ++nt) acc[nt] = vzero;
#pragma unroll
    for (int kk = 0; kk < 16; ++kk)
#pragma unroll
      for (int nt = 0; nt < 4; ++nt)
        acc[nt] = wmma_f32(af[kk], bf[kk][nt], acc[nt]);

    const int e0 = tile * 16;
    int sidx[8], didx[8];
#pragma unroll
    for (int r = 0; r < 8; ++r) {
      int ed = e0 + hv * 8 + r;
      sidx[r] = src[ed];
      didx[r] = dst[ed];
    }
#pragma unroll
    for (int nt = 0; nt < 4; ++nt) {
      const int col = nt * 16 + mlo;
      float ssum = 0.0f, ssq = 0.0f;
#pragma unroll
      for (int r = 0; r < 8; ++r) {
        const int s = sidx[r], d2 = didx[r];
        float z = acc[nt][r] + cb[nt] + Dx[(size_t)s * DD + col] +
                  Ex[(size_t)d2 * DD + col];
        float sg = sigmoidf_fast(z);
        atomicAdd(&num[(size_t)d2 * DD + col], sg * Bx[(size_t)s * DD + col]);
        atomicAdd(&den[(size_t)d2 * DD + col], sg);
        ssum += sg;
        ssq += sg * sg;
      }
      atomicAdd(&sstat[col], ssum);
      atomicAdd(&sstat[DD + col], ssq);
    }
  }
  __syncthreads();
  atomicAdd(&stats[128 + t], sstat[t]);
}

// ------------------------------------- K3: x-pre batchnorm statistics reduce
__global__ void __launch_bounds__(256)
xpre_stats_k(const float* __restrict__ Ax, const float* __restrict__ num,
             const float* __restrict__ den, float* __restrict__ stats) {
  __shared__ float sstat[2 * DD];
  const int t = threadIdx.x;
  if (t < 2 * DD) sstat[t] = 0.0f;
  __syncthreads();
  const size_t total = (size_t)NN * DD;
  size_t i = (size_t)blockIdx.x * 256 + t;
  const size_t stride = (size_t)gridDim.x * 256;  // multiple of 64 -> col fixed
  const int col = (int)(i & 63);
  float s = 0.0f, s2 = 0.0f;
  for (; i < total; i += stride) {
    float v = Ax[i] + num[i] / (den[i] + 1e-6f);
    s += v;
    s2 += v * v;
  }
  atomicAdd(&sstat[col], s);
  atomicAdd(&sstat[DD + col], s2);
  __syncthreads();
  if (t < 2 * DD) atomicAdd(&stats[t], sstat[t]);
}

// ----------------------------------------------------- K4: finalize mu / inv
__global__ void __launch_bounds__(128) finalize_stats_k(float* __restrict__ stats) {
  const int t = threadIdx.x;
  if (t < DD) {
    float mu = stats[t] * (1.0f / (float)NN);
    float var = stats[DD + t] * (1.0f / (float)NN) - mu * mu;
    stats[256 + t] = mu;
    stats[320 + t] = rsqrtf(var + 1e-5f);
  } else {
    int c = t - DD;
    float mu = stats[128 + c] * (1.0f / (float)EE);
    float var = stats[192 + c] * (1.0f / (float)EE) - mu * mu;
    stats[384 + c] = mu;
    stats[448 + c] = rsqrtf(var + 1e-5f);
  }
}

// ------------------------------------------------------------ K5: x output
__global__ void __launch_bounds__(256)
x_out_k(const float* __restrict__ x, const float* __restrict__ Ax,
        const float* __restrict__ num, const float* __restrict__ den,
        const float* __restrict__ gamma, const float* __restrict__ beta,
        const float* __restrict__ stats, float* __restrict__ out) {
  const size_t total = (size_t)NN * DD;
  size_t i = (size_t)blockIdx.x * 256 + threadIdx.x;
  const size_t stride = (size_t)gridDim.x * 256;
  for (; i < total; i += stride) {
    int col = (int)(i & 63);
    float v = Ax[i] + num[i] / (den[i] + 1e-6f);
    float z = (v - stats[256 + col]) * stats[320 + col] * gamma[col] + beta[col];
    out[i] = x[i] + fmaxf(z, 0.0f);
  }
}

// --------------------------- K6: edge pass 2 (recompute sigma, BN, residual)
__global__ void __launch_bounds__(128)
edge_pass2_k(const float* __restrict__ e, const int* __restrict__ src,
             const int* __restrict__ dst, const float* __restrict__ WC,
             const float* __restrict__ bC, const float* __restrict__ Dx,
             const float* __restrict__ Ex, const float* __restrict__ gamma,
             const float* __restrict__ beta, const float* __restrict__ stats,
             float* __restrict__ out_e) {
  __shared__ __align__(16) float etile[4][2][TILEF];
  __shared__ float wt[DD * DD];
  const int t = threadIdx.x;
  {
    int n = t >> 1, kb = (t & 1) * 32;
#pragma unroll
    for (int j = 0; j < 32; ++j) wt[n * DD + kb + j] = WC[(kb + j) * DD + n];
  }
  __syncthreads();

  const int wid = t >> 5, lane = t & 31;
  const int mlo = lane & 15, hv = lane >> 4, kh = hv * 2;
  const v8f vzero = {0.f, 0.f, 0.f, 0.f, 0.f, 0.f, 0.f, 0.f};

  v2f bf[16][4];
#pragma unroll
  for (int kk = 0; kk < 16; ++kk)
#pragma unroll
    for (int nt = 0; nt < 4; ++nt)
      bf[kk][nt] = *(const v2f*)&wt[(nt * 16 + mlo) * DD + kk * 4 + kh];
  float cb[4], mu[4], iv[4], gm[4], bt[4];
#pragma unroll
  for (int nt = 0; nt < 4; ++nt) {
    int col = nt * 16 + mlo;
    cb[nt] = bC[col];
    mu[nt] = stats[384 + col];
    iv[nt] = stats[448 + col];
    gm[nt] = gamma[col];
    bt[nt] = beta[col];
  }

  const float* p0 = &etile[wid][0][0];
  const float* p1 = &etile[wid][1][0];
  const uint32_t l0 = lds_off(p0), l1 = lds_off(p1);

  const int nTiles = EE / 16;
  const int step = gridDim.x * 4;
  int tile = blockIdx.x * 4 + wid;
  int b = 0;
  if (tile < nTiles) stage_tile_async(e + (size_t)tile * 16 * DD, l0, lane);
  for (; tile < nTiles; tile += step, b ^= 1) {
    const int nxt = tile + step;
    if (nxt < nTiles) {
      stage_tile_async(e + (size_t)nxt * 16 * DD, b ? l0 : l1, lane);
      asm volatile("s_wait_asynccnt 8" ::: "memory");
    } else {
      asm volatile("s_wait_asynccnt 0" ::: "memory");
    }
    const float* my = b ? p1 : p0;

    v2f af[16];
#pragma unroll
    for (int kk = 0; kk < 16; ++kk)
      af[kk] = *(const v2f*)&my[mlo * ROWP + kk * 4 + kh];

    v8f acc[4];
#pragma unroll
    for (int nt = 0; nt < 4; ++nt) acc[nt] = vzero;
#pragma unroll
    for (int kk = 0; kk < 16; ++kk)
#pragma unroll
      for (int nt = 0; nt < 4; ++nt)
        acc[nt] = wmma_f32(af[kk], bf[kk][nt], acc[nt]);

    const int e0 = tile * 16;
    int sidx[8], didx[8];
#pragma unroll
    for (int r = 0; r < 8; ++r) {
      int ed = e0 + hv * 8 + r;
      sidx[r] = src[ed];
      didx[r] = dst[ed];
    }
#pragma unroll
    for (int nt = 0; nt < 4; ++nt) {
      const int col = nt * 16 + mlo;
#pragma unroll
      for (int r = 0; r < 8; ++r) {
        const int ed = e0 + hv * 8 + r;
        float z = acc[nt][r] + cb[nt] + Dx[(size_t)sidx[r] * DD + col] +
                  Ex[(size_t)didx[r] * DD + col];
        float sg = sigmoidf_fast(z);
        float v = (sg - mu[nt]) * iv[nt] * gm[nt] + bt[nt];
        float ein = my[(hv * 8 + r) * ROWP + col];  // residual from staged tile
        __builtin_nontemporal_store(ein + fmaxf(v, 0.0f),
                                    &out_e[(size_t)ed * DD + col]);
      }
    }
  }
}

// ---------------------------------------------------------------- dispatch
extern "C" void kernel_launch(void* const* d_in, const int* in_sizes, int n_in,
                              void* d_out, int out_size, void* d_ws,
                              size_t ws_size, hipStream_t stream) {
  const float* x = (const float*)d_in[0];
  const float* e = (const float*)d_in[1];
  const int* src = (const int*)d_in[2];
  const int* dst = (const int*)d_in[3];
  const float* WA = (const float*)d_in[4];
  const float* bA = (const float*)d_in[5];
  const float* WB = (const float*)d_in[6];
  const float* bB = (const float*)d_in[7];
  const float* WC = (const float*)d_in[8];
  const float* bC = (const float*)d_in[9];
  const float* WDm = (const float*)d_in[10];
  const float* bD = (const float*)d_in[11];
  const float* WEm = (const float*)d_in[12];
  const float* bE = (const float*)d_in[13];
  const float* gamma_x = (const float*)d_in[14];
  const float* beta_x = (const float*)d_in[15];
  const float* gamma_e = (const float*)d_in[16];
  const float* beta_e = (const float*)d_in[17];

  float* ws = (float*)d_ws;
  const size_t ND = (size_t)NN * DD;
  float* Ax = ws;
  float* Bx = ws + ND;
  float* Dx = ws + 2 * ND;
  float* Ex = ws + 3 * ND;
  float* num = ws + 4 * ND;
  float* den = ws + 5 * ND;
  float* stats = ws + 6 * ND;  // [0:128) x sums, [128:256) e sums, [256:512) mu/inv

  float* out_x = (float*)d_out;
  float* out_e = out_x + ND;

  zero_ws_k<<<2048, 256, 0, stream>>>(num, 2 * ND + 512);
  node_gemm4_k<<<(NN / 16) / 8, 256, 0, stream>>>(x, WA, bA, WB, bB, WDm, bD,
                                                  WEm, bE, Ax, Bx, Dx, Ex);
  edge_pass1_k<<<4000, 128, 0, stream>>>(e, src, dst, WC, bC, Bx, Dx, Ex, num,
                                         den, stats);
  xpre_stats_k<<<512, 256, 0, stream>>>(Ax, num, den, stats);
  finalize_stats_k<<<1, 128, 0, stream>>>(stats);
  x_out_k<<<1024, 256, 0, stream>>>(x, Ax, num, den, gamma_x, beta_x, stats,
                                    out_x);
  edge_pass2_k<<<4000, 128, 0, stream>>>(e, src, dst, WC, bC, Dx, Ex, gamma_e,
                                         beta_e, stats, out_e);
}